// FinalLayer_11536282157398
// MI455X (gfx1250) — compile-verified
//
#include <hip/hip_runtime.h>
#include <math.h>

// Shapes from the reference
#define B_    8
#define T_    8192
#define D_    1024
#define TWO_D 2048
#define OUT_  3
#define EPS_  1e-6f

typedef float v2f __attribute__((ext_vector_type(2)));
typedef float v8f __attribute__((ext_vector_type(8)));

__device__ __forceinline__ float silu_f(float v) {
    return v * (1.0f / (1.0f + __expf(-v)));
}

// ---------------------------------------------------------------------------
// Kernel 1: mod[B,2D] = silu(cond) @ w_ada + b_ada
// f32 WMMA (V_WMMA_F32_16X16X4_F32), one wave per 16-wide N tile.
// grid = 16 blocks x 256 threads (8 waves); wave w handles n0 = (blk*8+w)*16.
//
// A-matrix (16x4 f32, 2 VGPRs): lane l (0-15) = row M=l, half h:
//   v0 = A[M][k+2h], v1 = A[M][k+2h+1].
// Key trick: D row M depends only on A row M, and we only store M=0..7,
// so lanes with l>=8 just re-read row (l&7) — broadcast, no masking, no
// exec-mask churn around the LDS loads.
// B-matrix (4x16) mirrored: lane l = col N=n0+l, v0 = B[k+2h][N], v1 = ..+1.
// C/D: VGPR r, lanes 0-15 -> (M=r, N=n0+lane)  => rows 0..7 live in low half.
// ---------------------------------------------------------------------------
__global__ __launch_bounds__(256) void ada_mod_wmma(
    const float* __restrict__ cond, const float* __restrict__ w_ada,
    const float* __restrict__ b_ada, float* __restrict__ mod)
{
    // silu(cond) staged once; even stride 1026 keeps (row*1026+kk)*4 8B-aligned
    // for ds_load_b64, and bank = (2*row + kk) % 64 is distinct per real row.
    __shared__ __align__(16) float As[B_ * 1026];
    const int tid = threadIdx.x;
    for (int i = tid; i < B_ * D_; i += 256) {
        const int r = i >> 10, c = i & (D_ - 1);
        As[r * 1026 + c] = silu_f(cond[i]);
    }
    __syncthreads();

    const int wave = tid >> 5;
    const int lane = tid & 31;
    const int half = lane >> 4;            // which K pair this half-wave holds
    const int l    = lane & 15;            // row (A) / col (B) index in tile
    const int arow = (l & (B_ - 1)) * 1026; // rows 8..15 alias rows 0..7 (unused outputs)
    const int n0   = (blockIdx.x * 8 + wave) * 16;

    v8f acc = {};
    #pragma unroll 4
    for (int k = 0; k < D_; k += 4) {
        const int kk = k + 2 * half;
        const v2f a = *(const v2f*)(&As[arow + kk]);   // ds_load_b64
        v2f bf;
        bf.x = w_ada[(size_t)kk       * TWO_D + n0 + l];
        bf.y = w_ada[(size_t)(kk + 1) * TWO_D + n0 + l];
        // 8 args: (neg_a, A, neg_b, B, c_mod, C, reuse_a, reuse_b)
        acc = __builtin_amdgcn_wmma_f32_16x16x4_f32(
            false, a, false, bf, (short)0, acc, false, false);
    }

    if (half == 0) {                       // lanes 0-15 hold M=0..7 in vgprs 0..7
        const float bb = b_ada[n0 + l];
        #pragma unroll
        for (int r = 0; r < B_; ++r)
            mod[r * TWO_D + n0 + l] = acc[r] + bb;
    }
}

// ---------------------------------------------------------------------------
// Kernel 2: fused LayerNorm + AdaLN modulate + SiLU + [D x 3] projection.
// One 256-thread block per (b,t) row; x is read exactly once as float4s.
// This is the 256 MiB / 23.3 TB/s bandwidth-bound part (~11 us floor).
// ---------------------------------------------------------------------------
__device__ __forceinline__ float wave_sum(float v) {
    #pragma unroll
    for (int off = 16; off > 0; off >>= 1)
        v += __shfl_xor(v, off, 32);
    return v;
}

__global__ __launch_bounds__(256) void ln_mod_silu_proj(
    const float* __restrict__ x, const float* __restrict__ mod,
    const float* __restrict__ w_proj, const float* __restrict__ b_proj,
    float* __restrict__ out)
{
    __shared__ float red[24];
    const int row  = blockIdx.x;            // 0 .. B*T-1
    const int b    = row >> 13;             // row / T_
    const int tid  = threadIdx.x;
    const int wave = tid >> 5, lane = tid & 31;

    // one float4 per thread covers the whole 1024-wide row
    const float4 xv = ((const float4*)(x + (size_t)row * D_))[tid];
    float s  = xv.x + xv.y + xv.z + xv.w;
    float sq = xv.x * xv.x + xv.y * xv.y + xv.z * xv.z + xv.w * xv.w;
    s  = wave_sum(s);
    sq = wave_sum(sq);
    if (lane == 0) { red[wave] = s; red[8 + wave] = sq; }
    __syncthreads();
    float ts = 0.0f, tq = 0.0f;
    #pragma unroll
    for (int i = 0; i < 8; ++i) { ts += red[i]; tq += red[8 + i]; }
    const float mu  = ts * (1.0f / D_);
    const float var = tq * (1.0f / D_) - mu * mu;
    const float rs  = rsqrtf(var + EPS_);

    // scale/shift (L2-resident, 64 KB total) and w_proj slab (12 consecutive floats)
    const float4 sc4 = ((const float4*)(mod + b * TWO_D))[tid];
    const float4 sh4 = ((const float4*)(mod + b * TWO_D + D_))[tid];
    const float4 w0  = ((const float4*)w_proj)[tid * 3 + 0];
    const float4 w1  = ((const float4*)w_proj)[tid * 3 + 1];
    const float4 w2  = ((const float4*)w_proj)[tid * 3 + 2];

    const float xe[4]  = {xv.x,  xv.y,  xv.z,  xv.w};
    const float sce[4] = {sc4.x, sc4.y, sc4.z, sc4.w};
    const float she[4] = {sh4.x, sh4.y, sh4.z, sh4.w};
    const float wf[12] = {w0.x, w0.y, w0.z, w0.w,
                          w1.x, w1.y, w1.z, w1.w,
                          w2.x, w2.y, w2.z, w2.w};

    float a0 = 0.0f, a1 = 0.0f, a2 = 0.0f;
    #pragma unroll
    for (int e = 0; e < 4; ++e) {
        const float y = silu_f((xe[e] - mu) * rs * (1.0f + sce[e]) + she[e]);
        a0 += y * wf[e * 3 + 0];
        a1 += y * wf[e * 3 + 1];
        a2 += y * wf[e * 3 + 2];
    }
    a0 = wave_sum(a0);
    a1 = wave_sum(a1);
    a2 = wave_sum(a2);
    __syncthreads();                        // red[] reuse
    if (lane == 0) { red[wave] = a0; red[8 + wave] = a1; red[16 + wave] = a2; }
    __syncthreads();
    if (tid == 0) {
        float r0 = 0.0f, r1 = 0.0f, r2 = 0.0f;
        #pragma unroll
        for (int i = 0; i < 8; ++i) {
            r0 += red[i]; r1 += red[8 + i]; r2 += red[16 + i];
        }
        out[row * OUT_ + 0] = r0 + b_proj[0];
        out[row * OUT_ + 1] = r1 + b_proj[1];
        out[row * OUT_ + 2] = r2 + b_proj[2];
    }
}

// ---------------------------------------------------------------------------
extern "C" void kernel_launch(void* const* d_in, const int* in_sizes, int n_in,
                              void* d_out, int out_size, void* d_ws, size_t ws_size,
                              hipStream_t stream) {
    (void)in_sizes; (void)n_in; (void)out_size; (void)ws_size;
    const float* x      = (const float*)d_in[0];
    const float* cond   = (const float*)d_in[1];
    const float* w_ada  = (const float*)d_in[2];
    const float* b_ada  = (const float*)d_in[3];
    const float* w_proj = (const float*)d_in[4];
    const float* b_proj = (const float*)d_in[5];
    float*       out    = (float*)d_out;
    float*       mod    = (float*)d_ws;     // B*2D floats = 64 KB scratch

    ada_mod_wmma<<<dim3(16), dim3(256), 0, stream>>>(cond, w_ada, b_ada, mod);
    ln_mod_silu_proj<<<dim3(B_ * T_), dim3(256), 0, stream>>>(x, mod, w_proj, b_proj, out);
}